// TRMB_89240830476827
// MI455X (gfx1250) — compile-verified
//
#include <hip/hip_runtime.h>
#include <hip/hip_bf16.h>

// Problem constants (match reference)
#define B_N   2048
#define DIN_N 1024
#define M_N   128
#define D_N   512
#define TEMP_INV   2.0f      // 1/0.5
#define WRITE_RATE 0.5f
#define DECAY_L    0.01f
#define EPS_V      1e-8f
#define LN_EPS_V   1e-5f

typedef __attribute__((ext_vector_type(2))) float v2f;
typedef __attribute__((ext_vector_type(8))) float v8f;
typedef unsigned int u32x4 __attribute__((ext_vector_type(4)));
typedef int          i32x4 __attribute__((ext_vector_type(4)));
typedef int          i32x8 __attribute__((ext_vector_type(8)));

__device__ __forceinline__ float wave_sum32(float v) {
#pragma unroll
  for (int off = 16; off > 0; off >>= 1) v += __shfl_xor(v, off, 32);
  return v;
}
__device__ __forceinline__ float wave_max32(float v) {
#pragma unroll
  for (int off = 16; off > 0; off >>= 1) v = fmaxf(v, __shfl_xor(v, off, 32));
  return v;
}

// ---------------------------------------------------------------------------
// Kernel 1: z = rel @ Ww^T + bw ; z_t = z / (||z|| + eps)
// grid = B/16 blocks of 256 threads (8 waves). Block owns 16 rows x 512 cols.
// WMMA f32 16x16x4: wave w computes col tiles (w + 8j)*16, j = 0..3.
// ---------------------------------------------------------------------------
__global__ void trmb_zt_gemm(const float* __restrict__ rel,
                             const float* __restrict__ Ww,
                             const float* __restrict__ bw,
                             float* __restrict__ zt) {
  extern __shared__ float smem[];
  const int AS = DIN_N + 8;   // 1032 : LDS stride for A tile (bank-conflict pad)
  const int ZS = D_N + 8;     // 520  : LDS stride for z tile
  float* As = smem;
  const int t = threadIdx.x, lane = t & 31, wave = t >> 5;
  const int rowBase = blockIdx.x * 16;

  // Stage A tile: 16 x 1024 f32 (64 KB) via float4
  for (int i4 = t; i4 < 16 * (DIN_N / 4); i4 += 256) {
    const int m = i4 >> 8;                 // 256 float4 per row
    const int d = (i4 & 255) << 2;
    float4 v = *(const float4*)(rel + (size_t)(rowBase + m) * DIN_N + d);
    *(float4*)(As + m * AS + d) = v;
  }
  __syncthreads();

  const int m16 = lane & 15;
  const int koff = (lane >> 4) << 1;       // lanes 0-15: K=k0,k0+1 ; 16-31: +2,+3
  const float* Ap = As + m16 * AS;
  v8f acc[4];
#pragma unroll
  for (int j = 0; j < 4; ++j) {
    const int col = (wave + 8 * j) * 16 + m16;       // B[k][n] = Ww[n][k]
    const float* Bp = Ww + (size_t)col * DIN_N;
    v8f c = {0.f, 0.f, 0.f, 0.f, 0.f, 0.f, 0.f, 0.f};
#pragma unroll 4
    for (int k0 = 0; k0 < DIN_N; k0 += 4) {
      float2 av = *(const float2*)(Ap + k0 + koff);
      float2 bv = *(const float2*)(Bp + k0 + koff);
      v2f a, b;
      a.x = av.x; a.y = av.y;
      b.x = bv.x; b.y = bv.y;
      c = __builtin_amdgcn_wmma_f32_16x16x4_f32(false, a, false, b,
                                                (short)0, c, false, false);
    }
    acc[j] = c;
  }
  __syncthreads();                          // everyone done reading As

  // Spill C tiles (+bias) into LDS so each wave can see full rows for the norm
  float* Zs = smem;                         // reuse: 16 x 520
  const int rbase = (lane >> 4) << 3;       // lanes 0-15 -> rows 0-7, 16-31 -> 8-15
#pragma unroll
  for (int j = 0; j < 4; ++j) {
    const int col = (wave + 8 * j) * 16 + m16;
    const float bias = bw[col];
#pragma unroll
    for (int r = 0; r < 8; ++r)
      Zs[(r + rbase) * ZS + col] = acc[j][r] + bias;
  }
  __syncthreads();

  // Per-row L2 norm + write z_t (2 rows per wave)
  for (int r = wave * 2; r < wave * 2 + 2; ++r) {
    float ss = 0.f;
#pragma unroll
    for (int c = lane; c < D_N; c += 32) { float v = Zs[r * ZS + c]; ss += v * v; }
    ss = wave_sum32(ss);
    const float inv = 1.f / (sqrtf(ss) + EPS_V);
    float* outp = zt + (size_t)(rowBase + r) * D_N;
#pragma unroll
    for (int c = lane; c < D_N; c += 32) outp[c] = Zs[r * ZS + c] * inv;
  }
}

// ---------------------------------------------------------------------------
// Kernel 2: per-batch fused memory ops. One block per batch element.
// memory_bank[b] (128x512 f32) staged into the 320KB WGP LDS by the Tensor
// Data Mover (one tensor_load_to_lds DMA per block, TENSORcnt-tracked, with
// hardware pad: +4 DWORDs every 128 DWORDs -> LDS row stride 516).
// ---------------------------------------------------------------------------
__global__ void trmb_mem_kernel(const float* __restrict__ memg,
                                const float* __restrict__ tsg,
                                const float* __restrict__ ztg,
                                const int*   __restrict__ stepp,
                                float* __restrict__ out_dw,
                                float* __restrict__ out_mem,
                                float* __restrict__ out_ts,
                                float* __restrict__ ctx_raw) {
  extern __shared__ float smem[];
  const int BS = D_N + 4;                   // 516 : bank row stride (TDM pad)
  float* bank = smem;                       // 128 * 516
  float* zs   = smem + M_N * BS;            // 512
  float* lg   = zs + D_N;                   // 128
  float* red  = lg + M_N;                   // scratch
  const int b = blockIdx.x;
  const int t = threadIdx.x, lane = t & 31, wave = t >> 5;
  const float* memb = memg + (size_t)b * (M_N * D_N);
  const float step = (float)(*stepp);

  zs[t]       = ztg[(size_t)b * D_N + t];
  zs[t + 256] = ztg[(size_t)b * D_N + t + 256];

#if __has_builtin(__builtin_amdgcn_tensor_load_to_lds)
  // --- TDM path: one DMA of the whole 128x512 tile, padded into LDS ---
  if (t < 32) {                              // one wave issues the descriptor
    const unsigned long long ga = (unsigned long long)(uintptr_t)memb;
    const unsigned int lds_addr = (unsigned int)(uintptr_t)bank; // low 32 = LDS offset
    u32x4 g0;
    g0[0] = 1u;                              // count=1 (valid), no gather
    g0[1] = lds_addr;                        // lds_addr[31:0]
    g0[2] = (unsigned int)(ga & 0xFFFFFFFFu);          // global_addr[31:0]
    g0[3] = (unsigned int)((ga >> 32) & 0x01FFFFFFu)   // global_addr[56:32]
            | (2u << 30);                    // type = 2 ("image")
    i32x8 g1;
    g1[0] = (2 << 16)                        // data_size = 4B
          | (1 << 20)                        // pad_enable
          | (6 << 22)                        // pad_interval: 128 DWORDs
          | (3 << 25);                       // pad_amount: 4 DWORDs
    g1[1] = (D_N & 0xFFFF) << 16;            // tensor_dim0 = 512 (low 16)
    g1[2] = (M_N & 0xFFFF) << 16;            // tensor_dim1 = 128 (low 16)
    g1[3] = (D_N & 0xFFFF) << 16;            // tile_dim0 = 512
    g1[4] = (M_N & 0xFFFF);                  // tile_dim1 = 128, tile_dim2 = 0
    g1[5] = D_N;                             // tensor_dim0_stride = 512 (low 32)
    g1[6] = 0;                               // stride high / dim1_stride
    g1[7] = 0;
    i32x4 gz4 = {0, 0, 0, 0};                // groups 2/3 unused (2D tensor)
    i32x8 gz8 = {0, 0, 0, 0, 0, 0, 0, 0};    // trailing group (unused)
    __builtin_amdgcn_tensor_load_to_lds(g0, g1, gz4, gz4, gz8, 0);
    __builtin_amdgcn_s_wait_tensorcnt(0);
  }
#else
  // --- fallback: stage through VGPRs ---
  for (int i4 = t; i4 < M_N * (D_N / 4); i4 += 256) {
    const int m = i4 >> 7;
    const int d = (i4 & 127) << 2;
    float4 v = *(const float4*)(memb + (size_t)m * D_N + d);
    *(float4*)(bank + m * BS + d) = v;
  }
#endif
  __syncthreads();

  // logits[m] = (bank[m] . z_t) / T
  if (t < M_N) {
    const float* rowp = bank + t * BS;
    float dot = 0.f;
#pragma unroll 8
    for (int d = 0; d < D_N; ++d) dot += rowp[d] * zs[d];
    lg[t] = dot * TEMP_INV;
  }
  __syncthreads();
  if (t < 32) {
    float mx = fmaxf(fmaxf(lg[t], lg[t + 32]), fmaxf(lg[t + 64], lg[t + 96]));
    mx = wave_max32(mx);
    if (t == 0) red[0] = mx;
  }
  __syncthreads();
  if (t < M_N) lg[t] = __expf(lg[t] - red[0]);
  __syncthreads();
  if (t < 32) {
    float s = lg[t] + lg[t + 32] + lg[t + 64] + lg[t + 96];
    s = wave_sum32(s);
    if (t == 0) red[1] = s;
  }
  __syncthreads();
  if (t < M_N) {
    const float w   = lg[t] / red[1];                              // softmax
    const float dec = fmaxf(__expf(-DECAY_L * (step - tsg[(size_t)b * M_N + t])), EPS_V);
    lg[t] = w * dec;
  }
  __syncthreads();
  if (t < 32) {
    float s = lg[t] + lg[t + 32] + lg[t + 64] + lg[t + 96];
    s = wave_sum32(s);
    if (t == 0) red[2] = s;
  }
  __syncthreads();
  if (t < M_N) {
    const float dw = lg[t] / (red[2] + EPS_V);
    lg[t] = dw;
    out_dw[(size_t)b * M_N + t] = dw;
    out_ts[(size_t)b * M_N + t] = (1.f - dw) * tsg[(size_t)b * M_N + t] + dw * step;
  }
  __syncthreads();

  // context_raw[d] = sum_m dw[m] * bank[m][d]
  for (int d = t; d < D_N; d += 256) {
    float acc = 0.f;
#pragma unroll 8
    for (int m = 0; m < M_N; ++m) acc += lg[m] * bank[m * BS + d];
    ctx_raw[(size_t)b * D_N + d] = acc;
  }

  // gated write + per-row renorm of new_mem (16 rows per wave)
  float* outm = out_mem + (size_t)b * (M_N * D_N);
  for (int m = wave * 16; m < wave * 16 + 16; ++m) {
    const float wm = lg[m] * WRITE_RATE;
    const float km = 1.f - wm;
    const float* rowp = bank + m * BS;
    float vals[16];
    float ss = 0.f;
#pragma unroll
    for (int j = 0; j < 16; ++j) {
      const int d = lane + 32 * j;
      const float v = km * rowp[d] + wm * zs[d];
      vals[j] = v; ss += v * v;
    }
    ss = wave_sum32(ss);
    const float inv = 1.f / (sqrtf(ss) + EPS_V);
#pragma unroll
    for (int j = 0; j < 16; ++j)
      outm[(size_t)m * D_N + lane + 32 * j] = vals[j] * inv;
  }
}

// ---------------------------------------------------------------------------
// Kernel 3: context = LayerNorm(ctx_raw @ Wr^T + br) — WMMA f32 + fused LN.
// ---------------------------------------------------------------------------
__global__ void trmb_ctx_gemm_ln(const float* __restrict__ ctxraw,
                                 const float* __restrict__ Wr,
                                 const float* __restrict__ br,
                                 const float* __restrict__ gamma,
                                 const float* __restrict__ beta,
                                 float* __restrict__ ctx) {
  extern __shared__ float smem[];
  const int AS = D_N + 8;                   // 520
  float* As = smem;                         // 16 x 520
  float* Cs = smem + 16 * AS;               // 16 x 520
  const int t = threadIdx.x, lane = t & 31, wave = t >> 5;
  const int rowBase = blockIdx.x * 16;

  for (int i4 = t; i4 < 16 * (D_N / 4); i4 += 256) {
    const int m = i4 >> 7;
    const int d = (i4 & 127) << 2;
    float4 v = *(const float4*)(ctxraw + (size_t)(rowBase + m) * D_N + d);
    *(float4*)(As + m * AS + d) = v;
  }
  __syncthreads();

  const int m16 = lane & 15;
  const int koff = (lane >> 4) << 1;
  const float* Ap = As + m16 * AS;
  const int rbase = (lane >> 4) << 3;
#pragma unroll
  for (int j = 0; j < 4; ++j) {
    const int col = (wave + 8 * j) * 16 + m16;
    const float* Bp = Wr + (size_t)col * D_N;
    v8f c = {0.f, 0.f, 0.f, 0.f, 0.f, 0.f, 0.f, 0.f};
#pragma unroll 4
    for (int k0 = 0; k0 < D_N; k0 += 4) {
      float2 av = *(const float2*)(Ap + k0 + koff);
      float2 bv = *(const float2*)(Bp + k0 + koff);
      v2f a, b;
      a.x = av.x; a.y = av.y;
      b.x = bv.x; b.y = bv.y;
      c = __builtin_amdgcn_wmma_f32_16x16x4_f32(false, a, false, b,
                                                (short)0, c, false, false);
    }
    const float bias = br[col];
#pragma unroll
    for (int r = 0; r < 8; ++r)
      Cs[(r + rbase) * AS + col] = c[r] + bias;
  }
  __syncthreads();

  // Fused LayerNorm (2 rows per wave)
  for (int r = wave * 2; r < wave * 2 + 2; ++r) {
    float s = 0.f, ss = 0.f;
#pragma unroll
    for (int c = lane; c < D_N; c += 32) {
      const float v = Cs[r * AS + c];
      s += v; ss += v * v;
    }
    s = wave_sum32(s); ss = wave_sum32(ss);
    const float mean = s * (1.f / D_N);
    const float var  = ss * (1.f / D_N) - mean * mean;
    const float inv  = rsqrtf(var + LN_EPS_V);
    float* outp = ctx + (size_t)(rowBase + r) * D_N;
#pragma unroll
    for (int c = lane; c < D_N; c += 32)
      outp[c] = (Cs[r * AS + c] - mean) * inv * gamma[c] + beta[c];
  }
}

// ---------------------------------------------------------------------------
extern "C" void kernel_launch(void* const* d_in, const int* in_sizes, int n_in,
                              void* d_out, int out_size, void* d_ws, size_t ws_size,
                              hipStream_t stream) {
  const float* rel   = (const float*)d_in[0];
  const float* memb  = (const float*)d_in[1];
  const float* tsg   = (const float*)d_in[2];
  const float* Ww    = (const float*)d_in[3];
  const float* bw    = (const float*)d_in[4];
  const float* Wr    = (const float*)d_in[5];
  const float* br    = (const float*)d_in[6];
  const float* gamma = (const float*)d_in[7];
  const float* beta  = (const float*)d_in[8];
  const int*   stepp = (const int*)d_in[9];

  float* out = (float*)d_out;
  // Outputs concatenated flat in return order:
  float* o_ctx = out;                                          // [B, D]
  float* o_zt  = o_ctx + (size_t)B_N * D_N;                    // [B, D]
  float* o_dw  = o_zt  + (size_t)B_N * D_N;                    // [B, M]
  float* o_mem = o_dw  + (size_t)B_N * M_N;                    // [B, M, D]
  float* o_ts  = o_mem + (size_t)B_N * M_N * D_N;              // [B, M]

  float* ctx_raw = (float*)d_ws;                               // [B, D] scratch

  const size_t smem1 = (size_t)16 * (DIN_N + 8) * sizeof(float);             // ~66 KB
  const size_t smem2 = (size_t)(M_N * (D_N + 4) + D_N + M_N + 8) * sizeof(float); // ~267 KB
  const size_t smem3 = (size_t)2 * 16 * (D_N + 8) * sizeof(float);           // ~66.6 KB

  (void)hipFuncSetAttribute((const void*)trmb_zt_gemm,
                            hipFuncAttributeMaxDynamicSharedMemorySize, (int)smem1);
  (void)hipFuncSetAttribute((const void*)trmb_mem_kernel,
                            hipFuncAttributeMaxDynamicSharedMemorySize, (int)smem2);
  (void)hipFuncSetAttribute((const void*)trmb_ctx_gemm_ln,
                            hipFuncAttributeMaxDynamicSharedMemorySize, (int)smem3);

  trmb_zt_gemm<<<dim3(B_N / 16), dim3(256), smem1, stream>>>(rel, Ww, bw, o_zt);
  trmb_mem_kernel<<<dim3(B_N), dim3(256), smem2, stream>>>(
      memb, tsg, o_zt, stepp, o_dw, o_mem, o_ts, ctx_raw);
  trmb_ctx_gemm_ln<<<dim3(B_N / 16), dim3(256), smem3, stream>>>(
      ctx_raw, Wr, br, gamma, beta, o_ctx);
}